// VQCodebook_59794534695129
// MI455X (gfx1250) — compile-verified
//
#include <hip/hip_runtime.h>

typedef __attribute__((ext_vector_type(2))) float v2f;
typedef __attribute__((ext_vector_type(8))) float v8f;

#define VQ_D        128
#define VQ_K        1024
#define VQ_NROWS    (128 * 1024)
#define ROWS_PER_WG 128              // 8 waves * 16 rows
#define LDS_STRIDE  132              // 128 + 4 pad -> conflict-free ds_load_b64
#define TILE_FLOATS (16 * LDS_STRIDE)
#define NTILES      (VQ_K / 16)
#define Q_OFF       0
#define LOSS_OFF    ((size_t)VQ_NROWS * VQ_D)          // 16777216
#define IDX_OFF     (LOSS_OFF + 1)                     // 16777217

// Pass 1: per-code squared norms into d_ws; also zero the loss accumulator.
__global__ __launch_bounds__(256) void vq_cnorm_kernel(
    const float* __restrict__ cb, float* __restrict__ cnorm,
    float* __restrict__ loss_slot) {
  int k = blockIdx.x * 256 + threadIdx.x;
  if (k < VQ_K) {
    const float4* row = (const float4*)(cb + (size_t)k * VQ_D);
    float s = 0.f;
#pragma unroll
    for (int i = 0; i < VQ_D / 4; ++i) {
      float4 t = row[i];
      s += t.x * t.x + t.y * t.y + t.z * t.z + t.w * t.w;
    }
    cnorm[k] = s;
  }
  if (k == 0) *loss_slot = 0.f;
}

// Pass 2: fused WMMA scores + running argmin + codebook gather + loss.
__global__ __launch_bounds__(256) void vq_main_kernel(
    const float* __restrict__ x, const float* __restrict__ cb,
    const float* __restrict__ cnorm, float* __restrict__ out) {
  __shared__ float ldsB[2 * TILE_FLOATS];   // double-buffered 16x128 tile
  __shared__ float ldsCN[VQ_K];             // all code norms

  const int tid  = threadIdx.x;
  const int lane = tid & 31;
  const int wave = tid >> 5;
  const int half = lane >> 4;   // 0: K pair {0,1}, 1: K pair {2,3}
  const int ln   = lane & 15;   // A: row in tile / B: column (code) in tile
  const int row0 = blockIdx.x * ROWS_PER_WG + wave * 16;

  for (int i = tid; i < VQ_K; i += 256) ldsCN[i] = cnorm[i];

  // A-fragment resident in VGPRs: lane holds 64 floats of row (row0+ln):
  // dims {4kk + 2*half, 4kk + 2*half + 1} for kk = 0..31  (ISA 16x4 layout).
  float areg[64];
  {
    const float* arow = x + (size_t)(row0 + ln) * VQ_D + half * 2;
#pragma unroll
    for (int kk = 0; kk < 32; ++kk) {
      v2f t = *(const v2f*)(arow + 4 * kk);
      areg[2 * kk]     = t.x;
      areg[2 * kk + 1] = t.y;
    }
  }

  // Staging lane geometry (256 threads move one 16x128 tile, 8 floats each).
  const int sr = tid >> 4;            // code row within tile
  const int sc = (tid & 15) * 8;      // dim start
  const float* stage_src = cb + (size_t)sr * VQ_D + sc;
  float* stage_dst = ldsB + sr * LDS_STRIDE + sc;

  // Prologue: stage tile 0 into buffer 0.
  {
    float4 t0 = *(const float4*)(stage_src);
    float4 t1 = *(const float4*)(stage_src + 4);
    *(float4*)(stage_dst)     = t0;
    *(float4*)(stage_dst + 4) = t1;
  }
  __syncthreads();

  // Running (min score, argmin) for the 8 rows this lane covers in C/D layout.
  float best[8];
  int   bidx[8];
#pragma unroll
  for (int v = 0; v < 8; ++v) { best[v] = 3.4e38f; bidx[v] = 0; }

  for (int tile = 0; tile < NTILES; ++tile) {
    const int cur = tile & 1;

    // Issue next tile's global loads before compute (latency hidden by WMMAs).
    float4 n0, n1;
    if (tile + 1 < NTILES) {
      const float* nsrc = stage_src + (size_t)(tile + 1) * 16 * VQ_D;
      n0 = *(const float4*)(nsrc);
      n1 = *(const float4*)(nsrc + 4);
      if (tile + 2 < NTILES)
        __builtin_prefetch(nsrc + 16 * VQ_D, 0, 3);
    }

    // 32 x v_wmma_f32_16x16x4_f32 over 4 independent accumulator chains:
    // 3 independent instructions between each RAW on C -> no XDL stalls.
    v8f acc0 = {}, acc1 = {}, acc2 = {}, acc3 = {};
    const float* brow = ldsB + cur * TILE_FLOATS + ln * LDS_STRIDE + half * 2;
#pragma unroll
    for (int kk = 0; kk < 32; kk += 4) {
      v2f a0 = { areg[2 * kk + 0], areg[2 * kk + 1] };
      v2f b0 = *(const v2f*)(brow + 4 * kk);
      acc0 = __builtin_amdgcn_wmma_f32_16x16x4_f32(
          false, a0, false, b0, (short)0, acc0, false, false);
      v2f a1 = { areg[2 * kk + 2], areg[2 * kk + 3] };
      v2f b1 = *(const v2f*)(brow + 4 * kk + 4);
      acc1 = __builtin_amdgcn_wmma_f32_16x16x4_f32(
          false, a1, false, b1, (short)0, acc1, false, false);
      v2f a2 = { areg[2 * kk + 4], areg[2 * kk + 5] };
      v2f b2 = *(const v2f*)(brow + 4 * kk + 8);
      acc2 = __builtin_amdgcn_wmma_f32_16x16x4_f32(
          false, a2, false, b2, (short)0, acc2, false, false);
      v2f a3 = { areg[2 * kk + 6], areg[2 * kk + 7] };
      v2f b3 = *(const v2f*)(brow + 4 * kk + 12);
      acc3 = __builtin_amdgcn_wmma_f32_16x16x4_f32(
          false, a3, false, b3, (short)0, acc3, false, false);
    }

    // Store next tile into the other buffer. Safe with a single barrier:
    // the last readers of buf (cur^1) finished before the previous barrier.
    if (tile + 1 < NTILES) {
      float* ndst = stage_dst + (cur ^ 1) * TILE_FLOATS;
      *(float4*)(ndst)     = n0;
      *(float4*)(ndst + 4) = n1;
    }

    // score = ||c||^2 - 2 x.c  (||x||^2 is constant per row -> drop)
    const int   code = tile * 16 + ln;     // this lane's column = code
    const float cn   = ldsCN[code];
#pragma unroll
    for (int v = 0; v < 8; ++v) {
      float s = cn - 2.0f * ((acc0[v] + acc1[v]) + (acc2[v] + acc3[v]));
      if (s < best[v]) { best[v] = s; bidx[v] = code; }
    }

    __syncthreads();
  }

  // Cross-lane argmin over the 16 columns held in each half-wave.
  // After this, every lane in a half holds the final result for row
  // (v + 8*half); ties resolved toward the lower code index (matches argmin).
#pragma unroll
  for (int v = 0; v < 8; ++v) {
    float bv = best[v];
    int   bi = bidx[v];
#pragma unroll
    for (int off = 1; off < 16; off <<= 1) {
      float ov = __shfl_xor(bv, off, 32);
      int   oi = __shfl_xor(bi, off, 32);
      if (ov < bv || (ov == bv && oi < bi)) { bv = ov; bi = oi; }
    }
    best[v] = bv;
    bidx[v] = bi;
  }

  // Gather winning codebook rows -> quantized output; fused loss partial.
  float lsum = 0.f;
#pragma unroll
  for (int r = 0; r < 16; ++r) {
    const int code = __shfl(bidx[r & 7], (r < 8) ? 0 : 16, 32);
    const float4 c4 = *(const float4*)(cb + (size_t)code * VQ_D + 4 * lane);
    const size_t base = (size_t)(row0 + r) * VQ_D + 4 * lane;
    const float4 x4 = *(const float4*)(x + base);
    *(float4*)(out + Q_OFF + base) = c4;   // quantized_st == quantized (value)
    const float dx = c4.x - x4.x, dy = c4.y - x4.y;
    const float dz = c4.z - x4.z, dw = c4.w - x4.w;
    lsum += dx * dx + dy * dy + dz * dz + dw * dw;
    if (lane == 0) out[IDX_OFF + row0 + r] = (float)code;
  }

  // vq_loss = (1 + 0.25) * mean((q - x)^2)
#pragma unroll
  for (int off = 1; off < 32; off <<= 1) lsum += __shfl_xor(lsum, off, 32);
  if (lane == 0)
    atomicAdd(out + LOSS_OFF, lsum * (1.25f / (float)((size_t)VQ_NROWS * VQ_D)));
}

extern "C" void kernel_launch(void* const* d_in, const int* in_sizes, int n_in,
                              void* d_out, int out_size, void* d_ws, size_t ws_size,
                              hipStream_t stream) {
  (void)in_sizes; (void)n_in; (void)out_size; (void)ws_size;
  const float* x   = (const float*)d_in[0];   // [128,1024,128] f32
  const float* cb  = (const float*)d_in[1];   // [1024,128] f32
  float* out       = (float*)d_out;           // q(16777216) + loss(1) + idx(131072)
  float* cnorm     = (float*)d_ws;            // 1024 floats scratch

  vq_cnorm_kernel<<<(VQ_K + 255) / 256, 256, 0, stream>>>(cb, cnorm, out + LOSS_OFF);
  vq_main_kernel<<<VQ_NROWS / ROWS_PER_WG, 256, 0, stream>>>(x, cb, cnorm, out);
}